// HybridSamplerQNN_65481071403988
// MI455X (gfx1250) — compile-verified
//
#include <hip/hip_runtime.h>
#include <hip/hip_bf16.h>

typedef float v2f __attribute__((ext_vector_type(2)));
typedef float v8f __attribute__((ext_vector_type(8)));

#define WAVE   32
#define ITERS  8          // tiles of 32 elements per wave
#define BLOCK  256        // 8 waves per block -> 2048 elements per block

// Native gfx1250 v_tanh_f32 when available; branch-free exp fallback otherwise
// (fallback is also what the host-side semantic pass parses).
__device__ __forceinline__ float fast_tanh(float x) {
#if __has_builtin(__builtin_amdgcn_tanhf)
    return __builtin_amdgcn_tanhf(x);
#else
    const float xc = fminf(fmaxf(x, -44.0f), 44.0f);   // exp(88) finite in f32
    const float t  = __expf(2.0f * xc);
    return (t - 1.0f) / (t + 1.0f);
#endif
}

struct QnnConsts {
    float b10, b11, b12, b13;                       // enc_b1
    float w200, w201, w202, w203;                   // enc_w2 row 0
    float w210, w211, w212, w213;                   // enc_w2 row 1
    float b20, b21;                                 // enc_b2
    float CW, SW;                                   // cos/sin(sum(weights)/2)
    float cw0, cb0;                                 // classifier
};

// One 32-element tile. Encoder layer-1 (x @ enc_w1^T) via V_WMMA_F32_16X16X4_F32:
//   A = enc_w1 zero-padded to 16x4 (A1: rows 0..3, A2: rows 8..11)
//   B = x^T tile (4x16): lanes 0..15 carry K=0..1 (x.x, x.y); K=2..3 zero.
// D layout (VGPR r: M=r on lanes 0-15, M=8+r on lanes 16-31) means every lane
// ends up with the 4 hidden pre-activations of element base+lane in D[0..3].
template<bool GUARD>
__device__ __forceinline__ void qnn_tile(const float2* __restrict__ xp,
                                         float* __restrict__ out,
                                         int base, int lane,
                                         v2f A1, v2f A2,
                                         const QnnConsts& k, int n)
{
    float e0x = 0.f, e0y = 0.f, e1x = 0.f, e1y = 0.f;
    if (lane < 16) {
        if (!GUARD || base + lane < n)      { float2 v = xp[base + lane];      e0x = v.x; e0y = v.y; }
        if (!GUARD || base + 16 + lane < n) { float2 v = xp[base + 16 + lane]; e1x = v.x; e1y = v.y; }
    }
    const v2f B1 = { e0x, e0y };   // elements base+0..15   (N = lane)
    const v2f B2 = { e1x, e1y };   // elements base+16..31

    const v8f acc = {0.f,0.f,0.f,0.f,0.f,0.f,0.f,0.f};
    v8f D1 = __builtin_amdgcn_wmma_f32_16x16x4_f32(false, A1, false, B1,
                                                   (short)0, acc, false, false);
    v8f D2 = __builtin_amdgcn_wmma_f32_16x16x4_f32(false, A2, false, B2,
                                                   (short)0, acc, false, false);

    float p0, p1, p2, p3;
    if (lane < 16) { p0 = D1[0]; p1 = D1[1]; p2 = D1[2]; p3 = D1[3]; }
    else           { p0 = D2[0]; p1 = D2[1]; p2 = D2[2]; p3 = D2[3]; }

    // encoder: tanh + Linear(4,2)
    const float h0 = fast_tanh(p0 + k.b10);
    const float h1 = fast_tanh(p1 + k.b11);
    const float h2 = fast_tanh(p2 + k.b12);
    const float h3 = fast_tanh(p3 + k.b13);
    const float q0 = k.w200*h0 + k.w201*h1 + k.w202*h2 + k.w203*h3 + k.b20;
    const float q1 = k.w210*h0 + k.w211*h1 + k.w212*h2 + k.w213*h3 + k.b21;

    // 2-qubit circuit (Ry's are real -> state stays real)
    const float c0 = __cosf(0.5f*q0), s0 = __sinf(0.5f*q0);
    const float c1 = __cosf(0.5f*q1), s1 = __sinf(0.5f*q1);
    // |psi> = Ry(q0) (x) Ry(q1) |00>, then CX (row1 entries swapped)
    const float r00 = c0*c1, r01 = c0*s1;
    const float r10 = s0*s1, r11 = s0*c1;
    // four Ry(weights[i]) on qubit0 compose to Ry(sum w)
    const float n00 = k.CW*r00 - k.SW*r10, n01 = k.CW*r01 - k.SW*r11;
    const float n10 = k.SW*r00 + k.CW*r10, n11 = k.SW*r01 + k.CW*r11;
    // trailing CX + qubit1 Ry's are row-orthogonal -> <Z0> unchanged
    const float E = (n00*n00 + n01*n01) - (n10*n10 + n11*n11);

    if (!GUARD || base + lane < n)
        out[base + lane] = E * k.cw0 + k.cb0;
}

__global__ void __launch_bounds__(BLOCK)
hybrid_qnn_kernel(const float2* __restrict__ xp,
                  const float*  __restrict__ enc_w1,   // (4,2) row-major
                  const float*  __restrict__ enc_b1,   // (4,)
                  const float*  __restrict__ enc_w2,   // (2,4) row-major
                  const float*  __restrict__ enc_b2,   // (2,)
                  const float*  __restrict__ wts,      // (4,)
                  const float*  __restrict__ cls_w,    // (1,1)
                  const float*  __restrict__ cls_b,    // (1,)
                  float*        __restrict__ out,      // (B,)
                  int n)
{
    const int lane  = threadIdx.x & (WAVE - 1);
    const int wave  = (int)((blockIdx.x * BLOCK + threadIdx.x) >> 5);
    const int wbase = wave * (WAVE * ITERS);

    QnnConsts k;
    k.b10 = enc_b1[0]; k.b11 = enc_b1[1]; k.b12 = enc_b1[2]; k.b13 = enc_b1[3];
    k.w200 = enc_w2[0]; k.w201 = enc_w2[1]; k.w202 = enc_w2[2]; k.w203 = enc_w2[3];
    k.w210 = enc_w2[4]; k.w211 = enc_w2[5]; k.w212 = enc_w2[6]; k.w213 = enc_w2[7];
    k.b20 = enc_b2[0]; k.b21 = enc_b2[1];
    const float hw = 0.5f * (wts[0] + wts[1] + wts[2] + wts[3]);
    k.CW = __cosf(hw); k.SW = __sinf(hw);
    k.cw0 = cls_w[0]; k.cb0 = cls_b[0];

    // A matrices (lanes 16..31 are K=2,3 -> zero padding)
    float a1k0 = 0.f, a1k1 = 0.f, a2k0 = 0.f, a2k1 = 0.f;
    if (lane < 4)               { a1k0 = enc_w1[2*lane];     a1k1 = enc_w1[2*lane+1]; }
    if (lane >= 8 && lane < 12) { a2k0 = enc_w1[2*(lane-8)]; a2k1 = enc_w1[2*(lane-8)+1]; }
    const v2f A1 = { a1k0, a1k1 };   // enc_w1 at rows M=0..3
    const v2f A2 = { a2k0, a2k1 };   // enc_w1 at rows M=8..11

    if (wbase + WAVE * ITERS <= n) {
        // fast path: whole 256-element wave range in bounds, no per-lane guards
        for (int it = 0; it < ITERS; ++it)
            qnn_tile<false>(xp, out, wbase + it * WAVE, lane, A1, A2, k, n);
    } else {
        for (int it = 0; it < ITERS; ++it) {
            const int base = wbase + it * WAVE;   // wave-uniform
            if (base >= n) break;
            qnn_tile<true>(xp, out, base, lane, A1, A2, k, n);
        }
    }
}

extern "C" void kernel_launch(void* const* d_in, const int* in_sizes, int n_in,
                              void* d_out, int out_size, void* d_ws, size_t ws_size,
                              hipStream_t stream) {
    (void)n_in; (void)d_ws; (void)ws_size; (void)out_size;
    const float2* xp    = (const float2*)d_in[0];
    const float* enc_w1 = (const float*)d_in[1];
    const float* enc_b1 = (const float*)d_in[2];
    const float* enc_w2 = (const float*)d_in[3];
    const float* enc_b2 = (const float*)d_in[4];
    const float* wts    = (const float*)d_in[5];
    const float* cls_w  = (const float*)d_in[6];
    const float* cls_b  = (const float*)d_in[7];
    float* out = (float*)d_out;

    const int n = in_sizes[0] / 2;                               // batch size
    const int elems_per_block = (BLOCK / WAVE) * WAVE * ITERS;   // 2048
    const int grid = (n + elems_per_block - 1) / elems_per_block;

    hybrid_qnn_kernel<<<grid, BLOCK, 0, stream>>>(
        xp, enc_w1, enc_b1, enc_w2, enc_b2, wts, cls_w, cls_b, out, n);
}